// WMeasure_Rot_6734508720664
// MI455X (gfx1250) — compile-verified
//
#include <hip/hip_runtime.h>
#include <cstdint>

#ifndef __has_builtin
#define __has_builtin(x) 0
#endif

#define TOTAL_ELEMS (1u << 25)   // 4096 * 8192
#define HALF_ELEMS  (1u << 24)
#define TPB   256
#define VEC   4
#define TILE  (TPB * VEC)                 // 1024 f32 per stream per stage (4 KB)
#define NSTAGES (HALF_ELEMS / TILE)       // 16384
#define GRID  1024

typedef unsigned int u32x4 __attribute__((ext_vector_type(4)));
typedef int          i32x8 __attribute__((ext_vector_type(8)));
typedef int          i32x4 __attribute__((ext_vector_type(4)));
typedef float        f32x4 __attribute__((ext_vector_type(4)));

#if defined(__HIP_DEVICE_COMPILE__) && \
    __has_builtin(__builtin_amdgcn_tensor_load_to_lds) && \
    __has_builtin(__builtin_amdgcn_s_wait_tensorcnt)
#define USE_TDM 1
#else
#define USE_TDM 0
#endif

// ---------------- Threefry-2x32, key = (0, 42)  (jax.random.key(42)) --------
__device__ __forceinline__ unsigned rotl32(unsigned x, int r) {
    return (x << r) | (x >> (32 - r));
}

__device__ __forceinline__ void threefry2x32_k42(unsigned c0, unsigned c1,
                                                 unsigned& o0, unsigned& o1) {
    const unsigned ks0 = 0u, ks1 = 42u, ks2 = 0x1BD11BDAu ^ 42u;
    unsigned x0 = c0 + ks0, x1 = c1 + ks1;
#define TF_R(r) { x0 += x1; x1 = rotl32(x1, (r)); x1 ^= x0; }
#define TF_G(a,b,c,d,ka,kb,i) TF_R(a) TF_R(b) TF_R(c) TF_R(d) x0 += (ka); x1 += (kb) + (i);
    TF_G(13,15,26, 6, ks1, ks2, 1u)
    TF_G(17,29,16,24, ks2, ks0, 2u)
    TF_G(13,15,26, 6, ks0, ks1, 3u)
    TF_G(17,29,16,24, ks1, ks2, 4u)
    TF_G(13,15,26, 6, ks2, ks0, 5u)
#undef TF_G
#undef TF_R
    o0 = x0; o1 = x1;
}

__device__ __forceinline__ float bits_to_uniform(unsigned b) {
    // matches jax: bitcast((b >> 9) | 0x3f800000) - 1.0  in [0, 1)
    return __uint_as_float((b >> 9) | 0x3F800000u) - 1.0f;
}

// ---------------- Per-element measurement update (native transcendentals) ---
__device__ __forceinline__ void wmeasure_elem(float xin, float ang, float lr,
                                              float u, float& meas, float& newang) {
    const float SIN_G = 0.70710678118654752f;      // sin(pi/4)
    const float SQ_P  = 0.54119610014619698f;      // sqrt(1 - sin(pi/4))
    const float SQ_M  = 1.30656296487637653f;      // sqrt(1 + sin(pi/4))
    const float PIO2  = 1.57079632679489662f;

    // tanh(2x) = 1 - 2/(exp(4x)+1); native v_exp/v_rcp, saturates correctly.
    float e4 = __expf(4.0f * xin);
    float th = __builtin_fmaf(-2.0f, __builtin_amdgcn_rcpf(e4 + 1.0f), 1.0f);
    float t  = __builtin_fmaf(PIO2, lr - th, ang);      // angles + theta
    float beta = __sinf(0.5f * t);                      // sin(t/2)
    float ez   = __cosf(t);                             // = cos^2(t/2)-sin^2(t/2)
    float eza  = ez * SIN_G;
    float p0   = __builtin_fmaf(0.5f, eza, 0.5f);       // p0/(p0+p1), p0+p1==1
    bool  hit  = (u < p0);
    float m    = hit ? 1.0f : -1.0f;
    float sqn  = hit ? SQ_P : SQ_M;                     // sqrt(1 - m*sin(g))
    float den  = __builtin_fmaf(m, eza, 1.0f);          // in [0.29, 1.71] > 0
    float side = beta * sqn * __builtin_amdgcn_rsqf(den);
    side = fminf(1.0f, fmaxf(-1.0f, side));
    // asin via A&S 4.4.46 minimax (|err| < 2e-8 rad): pi/2 - sqrt(1-x)*P(x)
    float ax = fabsf(side);
    float p  = -0.0012624911f;
    p = __builtin_fmaf(p, ax,  0.0066700901f);
    p = __builtin_fmaf(p, ax, -0.0170881256f);
    p = __builtin_fmaf(p, ax,  0.0308918810f);
    p = __builtin_fmaf(p, ax, -0.0501743046f);
    p = __builtin_fmaf(p, ax,  0.0889789874f);
    p = __builtin_fmaf(p, ax, -0.2145988016f);
    p = __builtin_fmaf(p, ax,  1.5707963050f);
    float r  = __builtin_amdgcn_sqrtf(1.0f - ax);
    float as = __builtin_fmaf(-r, p, PIO2);
    meas   = m;
    newang = 2.0f * copysignf(as, side);
}

__device__ __forceinline__ void compute_pair_vec4(
    const f32x4& inlo, const f32x4& anlo, const f32x4& lrlo,
    const f32x4& inhi, const f32x4& anhi, const f32x4& lrhi,
    unsigned ebase,
    f32x4& mlo, f32x4& mhi, f32x4& nalo, f32x4& nahi) {
#pragma unroll
    for (int k = 0; k < 4; ++k) {
        unsigned b0, b1;
        threefry2x32_k42(ebase + (unsigned)k, ebase + (unsigned)k + HALF_ELEMS, b0, b1);
        float u0 = bits_to_uniform(b0);
        float u1 = bits_to_uniform(b1);
        float m0, m1, a0, a1;
        wmeasure_elem(inlo[k], anlo[k], lrlo[k], u0, m0, a0);
        wmeasure_elem(inhi[k], anhi[k], lrhi[k], u1, m1, a1);
        mlo[k] = m0; nalo[k] = a0;
        mhi[k] = m1; nahi[k] = a1;
    }
}

// ---------------- TDM: 1D f32 tile, Global -> LDS ---------------------------
#if USE_TDM
__device__ __forceinline__ void tdm_load_f32_1d(const float* gsrc,
                                                unsigned lds_byte_off, int nelem) {
    unsigned long long ga = (unsigned long long)(uintptr_t)gsrc;
    u32x4 g0;
    g0[0] = 1u;                                            // count=1 (valid), user mode
    g0[1] = lds_byte_off;                                  // lds_addr [63:32]
    g0[2] = (unsigned)ga;                                  // global_addr [95:64]
    g0[3] = (unsigned)((ga >> 32) & 0x01FFFFFFu) | 0x80000000u; // addr[56:32] | type=2
    i32x8 g1;
    g1[0] = 0x00020000;                                    // wg_mask=0, data_size=2 (4B)
    g1[1] = (int)(((unsigned)nelem & 0xFFFFu) << 16);      // tensor_dim0[15:0] @ [63:48]
    g1[2] = (int)((((unsigned)nelem) >> 16) | (1u << 16)); // tensor_dim0[31:16] | tensor_dim1=1
    g1[3] = (int)(((unsigned)nelem & 0xFFFFu) << 16);      // tile_dim0 @ [127:112]
    g1[4] = 0;                                             // tile_dim1=0, tile_dim2=0 (1D)
    g1[5] = nelem;                                         // tensor_dim0_stride (unused, 1D)
    g1[6] = 0;
    g1[7] = 0;
    i32x4 z4 = {};
#if __clang_major__ >= 23
    i32x8 z8 = {};
    __builtin_amdgcn_tensor_load_to_lds(g0, g1, z4, z4, z8, 0);
#else
    __builtin_amdgcn_tensor_load_to_lds(g0, g1, z4, z4, 0);
#endif
}
#endif

// ---------------- Kernel -----------------------------------------------------
__global__ __launch_bounds__(TPB) void wmeasure_kernel(
    const float* __restrict__ xin, const float* __restrict__ ang,
    const float* __restrict__ lastr, float* __restrict__ out) {
    float* __restrict__ meas = out;
    float* __restrict__ nang = out + TOTAL_ELEMS;

#if USE_TDM
    __shared__ float ldsbuf[2][6][TILE];                   // 48 KB, double-buffered
    // ESCAPE: the TDM builtin writes LDS but takes only integer descriptor
    // words, so the compiler sees ldsbuf as never-stored and would fold the
    // ds_loads to undef. Hand the address to an opaque asm that "may write" it.
    {
        const float* lds_escape = &ldsbuf[0][0][0];
        asm volatile("" : : "s"(lds_escape) : "memory");
    }
    const int tid = threadIdx.x;
    // scalar (SGPR) wave-id test -> real s_cbranch; TDM ignores EXEC so only
    // wave 0 may reach the issue instructions.
    const bool wave0 = (__builtin_amdgcn_readfirstlane((int)threadIdx.x) == 0);

    auto issue_stage = [&](unsigned stage, int buf) {
        unsigned lo  = stage * (unsigned)TILE;
        unsigned off = (unsigned)buf * 6u * (TILE * 4u);   // byte offset, LDS base 0
        tdm_load_f32_1d(xin   + lo,              off + 0u * (TILE * 4u), TILE);
        tdm_load_f32_1d(ang   + lo,              off + 1u * (TILE * 4u), TILE);
        tdm_load_f32_1d(lastr + lo,              off + 2u * (TILE * 4u), TILE);
        tdm_load_f32_1d(xin   + lo + HALF_ELEMS, off + 3u * (TILE * 4u), TILE);
        tdm_load_f32_1d(ang   + lo + HALF_ELEMS, off + 4u * (TILE * 4u), TILE);
        tdm_load_f32_1d(lastr + lo + HALF_ELEMS, off + 5u * (TILE * 4u), TILE);
    };

    unsigned stage = blockIdx.x;
    if (stage < NSTAGES && wave0) issue_stage(stage, 0);   // prologue: 6 outstanding
    int buf = 0;
    for (; stage < NSTAGES; stage += gridDim.x, buf ^= 1) {
        unsigned next = stage + gridDim.x;
        if (next < NSTAGES) {
            if (wave0) issue_stage(next, buf ^ 1);         // 12 outstanding
            __builtin_amdgcn_s_wait_tensorcnt(6);          // stage `stage` landed (in-order)
        } else {
            __builtin_amdgcn_s_wait_tensorcnt(0);
        }
        __syncthreads();                                   // publish LDS to all waves
        // Opaque fence: ds_loads below must not be hoisted above the wait nor
        // folded (LDS was written behind the compiler's back).
        asm volatile("" : : : "memory");

        const float* b = &ldsbuf[buf][0][0];
        f32x4 inlo = *(const f32x4*)(b + 0 * TILE + tid * 4);
        f32x4 anlo = *(const f32x4*)(b + 1 * TILE + tid * 4);
        f32x4 lrlo = *(const f32x4*)(b + 2 * TILE + tid * 4);
        f32x4 inhi = *(const f32x4*)(b + 3 * TILE + tid * 4);
        f32x4 anhi = *(const f32x4*)(b + 4 * TILE + tid * 4);
        f32x4 lrhi = *(const f32x4*)(b + 5 * TILE + tid * 4);

        f32x4 mlo, mhi, nalo, nahi;
        unsigned e = stage * (unsigned)TILE + (unsigned)tid * 4u;
        compute_pair_vec4(inlo, anlo, lrlo, inhi, anhi, lrhi, e, mlo, mhi, nalo, nahi);

        // Outputs are a write-only 256 MB stream: non-temporal, keep L2 clean.
        __builtin_nontemporal_store(mlo,  (f32x4*)(meas + e));
        __builtin_nontemporal_store(mhi,  (f32x4*)(meas + e + HALF_ELEMS));
        __builtin_nontemporal_store(nalo, (f32x4*)(nang + e));
        __builtin_nontemporal_store(nahi, (f32x4*)(nang + e + HALF_ELEMS));

        __syncthreads();                                   // all done reading buf before reuse
    }
#else
    // Fallback: direct b128 streaming (still bandwidth-optimal, no TDM builtin).
    unsigned nvec = HALF_ELEMS / 4u;
    for (unsigned g = blockIdx.x * blockDim.x + threadIdx.x; g < nvec;
         g += gridDim.x * blockDim.x) {
        unsigned e = g * 4u;
        f32x4 inlo = *(const f32x4*)(xin   + e);
        f32x4 anlo = *(const f32x4*)(ang   + e);
        f32x4 lrlo = *(const f32x4*)(lastr + e);
        f32x4 inhi = *(const f32x4*)(xin   + e + HALF_ELEMS);
        f32x4 anhi = *(const f32x4*)(ang   + e + HALF_ELEMS);
        f32x4 lrhi = *(const f32x4*)(lastr + e + HALF_ELEMS);
        f32x4 mlo, mhi, nalo, nahi;
        compute_pair_vec4(inlo, anlo, lrlo, inhi, anhi, lrhi, e, mlo, mhi, nalo, nahi);
        __builtin_nontemporal_store(mlo,  (f32x4*)(meas + e));
        __builtin_nontemporal_store(mhi,  (f32x4*)(meas + e + HALF_ELEMS));
        __builtin_nontemporal_store(nalo, (f32x4*)(nang + e));
        __builtin_nontemporal_store(nahi, (f32x4*)(nang + e + HALF_ELEMS));
    }
#endif
}

extern "C" void kernel_launch(void* const* d_in, const int* in_sizes, int n_in,
                              void* d_out, int out_size, void* d_ws, size_t ws_size,
                              hipStream_t stream) {
    (void)in_sizes; (void)n_in; (void)out_size; (void)d_ws; (void)ws_size;
    const float* xin   = (const float*)d_in[0];
    const float* ang   = (const float*)d_in[1];
    const float* lastr = (const float*)d_in[2];
    float* out = (float*)d_out;
    wmeasure_kernel<<<GRID, TPB, 0, stream>>>(xin, ang, lastr, out);
}